// HierarchicalModel_15229954032058
// MI455X (gfx1250) — compile-verified
//
#include <hip/hip_runtime.h>
#include <math.h>

// ---------------------------------------------------------------------------
// Fused hierarchical MLP for MI455X (gfx1250, wave32, WMMA).
//
// Roofline: reading x (262144 x 512 f32 = 512MB) once at 23.3 TB/s ~= 22us
// is the floor; weights are <100KB (cache-resident), output is 5.25MB.
// One fused kernel: x read exactly once, f32 WMMA (v_wmma_f32_16x16x4_f32)
// for both GEMM layers, and a cheap closed-form mish (1x v_exp_f32 + rcp)
// so activation VALU cost (~6us) stays below the bandwidth floor.
// ---------------------------------------------------------------------------

typedef __attribute__((ext_vector_type(2))) float v2f;
typedef __attribute__((ext_vector_type(4))) float v4f;
typedef __attribute__((ext_vector_type(8))) float v8f;

static constexpr int B_ROWS = 262144;
static constexpr int F_IN   = 512;
static constexpr int H1     = 32;
static constexpr int H2     = 128;
static constexpr int NCLS   = 4;

static constexpr int WAVES          = 4;               // 4 waves / block
static constexpr int TPB            = WAVES * 32;      // 128 threads
static constexpr int ROWS_PER_WAVE  = 16;              // one WMMA M-tile
static constexpr int ROWS_PER_BLOCK = WAVES * ROWS_PER_WAVE; // 64

__device__ __forceinline__ v8f wmma_f32x4(v2f a, v2f b, v8f c) {
    // D = A(16x4,f32) * B(4x16,f32) + C(16x16,f32)
    return __builtin_amdgcn_wmma_f32_16x16x4_f32(
        /*neg_a=*/false, a, /*neg_b=*/false, b,
        /*c_mod=*/(short)0, c, /*reuse_a=*/false, /*reuse_b=*/false);
}

__device__ __forceinline__ float mishf(float v) {
    // mish(v) = v * tanh(softplus(v)).
    // Closed form: with e = exp(v), tanh(log(1+e)) = ((1+e)^2-1)/((1+e)^2+1),
    // so mish(v) = v * n / (n + 2) with n = e*(e+2).
    // Clamp the exp argument at 40: exp(40)^2 ~ 5.5e34 (no overflow) and the
    // ratio is exactly 1.0f there, so mish(v)=v for v>=40 as required.
    const float e = __expf(fminf(v, 40.0f));   // v_exp_f32 fast path
    const float n = e * (e + 2.0f);
    return v * (n / (n + 2.0f));               // v -> 0 as v -> -inf  (e -> 0)
}

__global__ __launch_bounds__(TPB) void fused_hier_mlp(
    const float* __restrict__ x,   const float* __restrict__ Win,
    const float* __restrict__ bin, const float* __restrict__ Wh,
    const float* __restrict__ bh,  const float* __restrict__ Wc,
    const float* __restrict__ bc,  const float* __restrict__ Wr,
    const float* __restrict__ br,
    float* __restrict__ out_logits, float* __restrict__ out_reg)
{
    __shared__ __align__(16) float sWc [H2 * NCLS];                 // [j][c]
    __shared__ __align__(16) float sWrT[H2 * NCLS];                 // Wr^T: [j][c]
    __shared__ __align__(16) float sH1 [WAVES][ROWS_PER_WAVE * H1]; // wave-private
    __shared__ __align__(16) float sH2 [WAVES][ROWS_PER_WAVE * H2]; // wave-private

    const int tid = threadIdx.x;

    // Stage the tiny head weights once per block (shared by all waves).
    for (int i = tid; i < H2 * NCLS; i += TPB) {
        sWc[i] = Wc[i];                    // Wc is [128][4] row-major already
        const int j = i >> 2, c = i & 3;
        sWrT[i] = Wr[c * H2 + j];          // transpose Wr[4][128] -> [128][4]
    }
    __syncthreads();

    const int wave = tid >> 5;
    const int lane = tid & 31;
    const int j16  = lane & 15;            // N-col / M-row within 16
    const int h16  = lane >> 4;            // which half of the wave
    const int row0 = blockIdx.x * ROWS_PER_BLOCK + wave * ROWS_PER_WAVE;

    // ---------------- Layer 1: h1 = mish(x @ Win + bin)   [16 x 32] -------
    // A frag (16x4 f32): lane row = j16; V0/V1 hold K = k0+2*h16, k0+1+2*h16.
    // B frag (4x16 f32): V0/V1 hold rows K = k0+2*h16, +1 at col n0+j16.
    v8f acc0 = {0.f,0.f,0.f,0.f,0.f,0.f,0.f,0.f};
    v8f acc1 = acc0;
    const float* xrow = x + (size_t)(row0 + j16) * F_IN;
    for (int k0 = 0; k0 < F_IN; k0 += 4) {
        const int kk = k0 + 2 * h16;
        const v2f afrag = *reinterpret_cast<const v2f*>(xrow + kk); // 8B aligned
        v2f b0, b1;
        b0.x = Win[ kk      * H1 + j16];
        b0.y = Win[(kk + 1) * H1 + j16];
        b1.x = Win[ kk      * H1 + 16 + j16];
        b1.y = Win[(kk + 1) * H1 + 16 + j16];
        acc0 = wmma_f32x4(afrag, b0, acc0);
        acc1 = wmma_f32x4(afrag, b1, acc1);
    }
    // D layout: acc[v] = h1[v + 8*h16][col], col = j16 (acc0) / 16+j16 (acc1)
    {
        const float bin0 = bin[j16], bin1 = bin[16 + j16];
        float* s1 = sH1[wave];
        #pragma unroll
        for (int v = 0; v < 8; ++v) {
            const int r = v + 8 * h16;
            s1[r * H1 + j16]      = mishf(acc0[v] + bin0);
            s1[r * H1 + 16 + j16] = mishf(acc1[v] + bin1);
        }
    }
    // Same-wave LDS RAW is ordered by hardware (DScnt, in-order) -- no barrier.

    // ---------------- Layer 2: h2 = mish(h1 @ Wh + bh)    [16 x 128] ------
    v8f acc[8];
    #pragma unroll
    for (int t = 0; t < 8; ++t) acc[t] = (v8f){0.f,0.f,0.f,0.f,0.f,0.f,0.f,0.f};
    {
        const float* s1 = sH1[wave];
        for (int k0 = 0; k0 < H1; k0 += 4) {
            const int ka = k0 + 2 * h16;
            const v2f afrag = *reinterpret_cast<const v2f*>(&s1[j16 * H1 + ka]);
            #pragma unroll
            for (int t = 0; t < 8; ++t) {
                const int n = t * 16 + j16;
                v2f bfrag;
                bfrag.x = Wh[ ka      * H2 + n];
                bfrag.y = Wh[(ka + 1) * H2 + n];
                acc[t] = wmma_f32x4(afrag, bfrag, acc[t]);
            }
        }
    }
    {
        float* s2 = sH2[wave];
        #pragma unroll
        for (int t = 0; t < 8; ++t) {
            const int col = t * 16 + j16;
            const float bb = bh[col];
            #pragma unroll
            for (int v = 0; v < 8; ++v) {
                const int r = v + 8 * h16;
                s2[r * H2 + col] = mishf(acc[t][v] + bb);
            }
        }
    }

    // ---------------- Heads: logits, per-class regressors, route ----------
    // lane -> (row = j16, cols [h16*64, h16*64+64)); combine halves via shfl.
    float lg[4] = {0.f,0.f,0.f,0.f};
    float rg[4] = {0.f,0.f,0.f,0.f};
    {
        const float* s2row = &sH2[wave][j16 * H2 + h16 * 64];
        const float* wcc   = &sWc [(h16 * 64) * 4];
        const float* wrr   = &sWrT[(h16 * 64) * 4];
        for (int j = 0; j < 64; ++j) {
            const float hv = s2row[j];
            const v4f wc = *reinterpret_cast<const v4f*>(wcc + j * 4);
            const v4f wr = *reinterpret_cast<const v4f*>(wrr + j * 4);
            #pragma unroll
            for (int c = 0; c < 4; ++c) {
                lg[c] = fmaf(hv, wc[c], lg[c]);
                rg[c] = fmaf(hv, wr[c], rg[c]);
            }
        }
    }
    #pragma unroll
    for (int c = 0; c < 4; ++c) {
        lg[c] += __shfl_xor(lg[c], 16, 32);
        rg[c] += __shfl_xor(rg[c], 16, 32);
    }

    if (h16 == 0) {
        const int row = row0 + j16;
        float lv[4];
        #pragma unroll
        for (int c = 0; c < 4; ++c) lv[c] = lg[c] + bc[c];

        // first-max argmax (matches jnp.argmax tie-breaking)
        int best = 0; float bv = lv[0];
        #pragma unroll
        for (int c = 1; c < 4; ++c) { if (lv[c] > bv) { bv = lv[c]; best = c; } }

        const float LOc[4] = {0.f, 2.f, 4.f, 6.f};
        const float HIc[4] = {2.f, 4.f, 6.f, 10.f};
        float rv[4];
        #pragma unroll
        for (int c = 0; c < 4; ++c)
            rv[c] = fminf(fmaxf(rg[c] + br[c], LOc[c]), HIc[c]);

        const v4f outv = {lv[0], lv[1], lv[2], lv[3]};
        *reinterpret_cast<v4f*>(&out_logits[(size_t)row * 4]) = outv;
        out_reg[row] = rv[best];
    }
}

extern "C" void kernel_launch(void* const* d_in, const int* in_sizes, int n_in,
                              void* d_out, int out_size, void* d_ws, size_t ws_size,
                              hipStream_t stream) {
    const float* x   = (const float*)d_in[0];
    const float* Win = (const float*)d_in[1];
    const float* bin = (const float*)d_in[2];
    const float* Wh  = (const float*)d_in[3];
    const float* bh  = (const float*)d_in[4];
    const float* Wc  = (const float*)d_in[5];
    const float* bc  = (const float*)d_in[6];
    const float* Wr  = (const float*)d_in[7];
    const float* br  = (const float*)d_in[8];

    float* out        = (float*)d_out;
    float* out_logits = out;                       // [B, 4]
    float* out_reg    = out + (size_t)B_ROWS * 4;  // [B]

    const dim3 grid(B_ROWS / ROWS_PER_BLOCK);
    fused_hier_mlp<<<grid, TPB, 0, stream>>>(
        x, Win, bin, Wh, bh, Wc, bc, Wr, br, out_logits, out_reg);
}